// RenderMeshColor_15401752723591
// MI455X (gfx1250) — compile-verified
//
#include <hip/hip_runtime.h>
#include <math.h>

typedef __attribute__((ext_vector_type(2))) float v2f;
typedef __attribute__((ext_vector_type(8))) float v8f;

#define NTS(p, v) __builtin_nontemporal_store((v), (p))

// ---------------------------------------------------------------------------
// Phase 1: camera transform of all B*P points via V_WMMA_F32_16X16X4_F32.
//   D = A(16x4) * B(4x16):
//     A rows 0..2 = [R row j | t_j] with t = -R*cam_pos  (rows 3..15 zero)
//     B cols 0..15 = 16 homogeneous points (x,y,z,1)
//   => D[j][n] = camera-space coord j of point n.
//   D layout (ISA 7.12.2): lane n (0..15), VGPR r -> element (M=r, N=n),
//   so lanes 0..15 read their point's x,y,z straight from d[0..2].
// Operand layouts (ISA 7.12.2):
//   A 16x4 : lane l, M=l%16; VGPR0 = A[M][l<16?0:2], VGPR1 = A[M][l<16?1:3]
//   B 4x16 : lane l, N=l%16; lanes 0-15 hold K=0,1; lanes 16-31 hold K=2,3
// Stores {x,y,z, 1/(z*proj2)} per point as an aligned float4 (L2-resident
// gather table for phase 2).
// ---------------------------------------------------------------------------
__global__ __launch_bounds__(256)
void transform_points_wmma(const float* __restrict__ pts_in,
                           const float* __restrict__ cam_rot,   // [B][3][3]
                           const float* __restrict__ cam_pos,   // [B][3]
                           const float* __restrict__ cam_proj,  // [3]
                           float* __restrict__ pts4_ws,         // [B][P][4]
                           int P)
{
    const int b    = blockIdx.y;
    const int lane = threadIdx.x & 31;
    const int wave = threadIdx.x >> 5;
    const int M    = lane & 15;
    const bool hi  = lane >= 16;

    // A operand: transform rows (rows >=3 are zero)
    float a0 = 0.0f, a1 = 0.0f;
    if (M < 3) {
        const float* R = cam_rot + b * 9 + M * 3;
        const float r0 = R[0], r1 = R[1], r2 = R[2];
        const float* C = cam_pos + b * 3;
        const float t  = -(r0 * C[0] + r1 * C[1] + r2 * C[2]);
        a0 = hi ? r2 : r0;   // K=0 / K=2
        a1 = hi ? t  : r1;   // K=1 / K=3
    }
    const float proj2 = cam_proj[2];

    // B operand: 16 points (columns); both half-waves load the same point
    const int p = blockIdx.x * 128 + wave * 16 + M;
    float x = 0.0f, y = 0.0f, z = 0.0f;
    if (p < P) {
        const float* q = pts_in + ((size_t)b * P + p) * 3;
        x = q[0]; y = q[1]; z = q[2];
    }

    v2f A;  A[0] = a0;           A[1] = a1;
    v2f Bv; Bv[0] = hi ? z : x;  Bv[1] = hi ? 1.0f : y;
    v8f c = {};
    // (neg_a, A, neg_b, B, c_mod, C, reuse_a, reuse_b)
    v8f d = __builtin_amdgcn_wmma_f32_16x16x4_f32(false, A, false, Bv,
                                                  (short)0, c, false, false);

    if (!hi && p < P) {
        const float X = d[0], Y = d[1], Z = d[2];
        const float w = 1.0f / (Z * proj2);     // reused for 2D projection
        float4 o; o.x = X; o.y = Y; o.z = Z; o.w = w;
        *(float4*)(pts4_ws + ((size_t)b * P + p) * 4) = o;
    }
}

// ---------------------------------------------------------------------------
// Phase 2: one thread per face; int64 indices kept in registers while looping
// all B batches (faces read once). Gathers hit the L2-resident pts4/tfcolor
// tables; all 28 output floats per (b,f) use non-temporal stores so the
// 358 MB output stream does not evict the gather tables from L2.
// ---------------------------------------------------------------------------
__global__ __launch_bounds__(256)
void assemble_faces(const float* __restrict__ pts4,      // [B][P][4] ws
                    const float* __restrict__ tfcolor,   // [B][P][3]
                    const long long* __restrict__ faces, // [F][3]
                    const float* __restrict__ cam_proj,  // [3]
                    float* __restrict__ out,
                    int P, int F, int B)
{
    const int f = blockIdx.x * blockDim.x + threadIdx.x;
    if (f >= F) return;

    const long long i0 = faces[(size_t)f * 3 + 0];
    const long long i1 = faces[(size_t)f * 3 + 1];
    const long long i2 = faces[(size_t)f * 3 + 2];
    const float pr0 = cam_proj[0], pr1 = cam_proj[1];

    const size_t BF = (size_t)B * F;
    float* o_p3 = out;                 // points3d_bxfx9
    float* o_p2 = o_p3 + BF * 9;       // points2d_bxfx6
    float* o_nz = o_p2 + BF * 6;       // normalz_bxfx1
    float* o_c9 = o_nz + BF;           // tfcolor_bxfx9
    float* o_n1 = o_c9 + BF * 9;       // normal1_bxfx3

    for (int b = 0; b < B; ++b) {
        const size_t pbase = (size_t)b * P;
        const float4 v0 = *(const float4*)(pts4 + (pbase + (size_t)i0) * 4);
        const float4 v1 = *(const float4*)(pts4 + (pbase + (size_t)i1) * 4);
        const float4 v2 = *(const float4*)(pts4 + (pbase + (size_t)i2) * 4);

        const float* c0 = tfcolor + (pbase + (size_t)i0) * 3;
        const float* c1 = tfcolor + (pbase + (size_t)i1) * 3;
        const float* c2 = tfcolor + (pbase + (size_t)i2) * 3;

        const size_t bf = (size_t)b * F + f;

        // points3d_bxfx9 = [pf0 | pf1 | pf2]
        {
            float* q = o_p3 + bf * 9;
            NTS(q + 0, v0.x); NTS(q + 1, v0.y); NTS(q + 2, v0.z);
            NTS(q + 3, v1.x); NTS(q + 4, v1.y); NTS(q + 5, v1.z);
            NTS(q + 6, v2.x); NTS(q + 7, v2.y); NTS(q + 8, v2.z);
        }
        // points2d_bxfx6 : u = x*p0/(z*p2), v = y*p1/(z*p2); v?.w = 1/(z*p2)
        {
            float* q = o_p2 + bf * 6;
            NTS(q + 0, v0.x * pr0 * v0.w); NTS(q + 1, v0.y * pr1 * v0.w);
            NTS(q + 2, v1.x * pr0 * v1.w); NTS(q + 3, v1.y * pr1 * v1.w);
            NTS(q + 4, v2.x * pr0 * v2.w); NTS(q + 5, v2.y * pr1 * v2.w);
        }
        // normal = cross(pf1-pf0, pf2-pf0)
        {
            const float e0x = v1.x - v0.x, e0y = v1.y - v0.y, e0z = v1.z - v0.z;
            const float e1x = v2.x - v0.x, e1y = v2.y - v0.y, e1z = v2.z - v0.z;
            const float nx = e0y * e1z - e0z * e1y;
            const float ny = e0z * e1x - e0x * e1z;
            const float nz = e0x * e1y - e0y * e1x;
            NTS(o_nz + bf, nz);
            const float len = sqrtf(nx * nx + ny * ny + nz * nz);
            const float inv = 1.0f / (len + 1e-15f);
            float* q = o_n1 + bf * 3;
            NTS(q + 0, nx * inv); NTS(q + 1, ny * inv); NTS(q + 2, nz * inv);
        }
        // tfcolor_bxfx9
        {
            float* q = o_c9 + bf * 9;
            NTS(q + 0, c0[0]); NTS(q + 1, c0[1]); NTS(q + 2, c0[2]);
            NTS(q + 3, c1[0]); NTS(q + 4, c1[1]); NTS(q + 5, c1[2]);
            NTS(q + 6, c2[0]); NTS(q + 7, c2[1]); NTS(q + 8, c2[2]);
        }
    }
}

extern "C" void kernel_launch(void* const* d_in, const int* in_sizes, int n_in,
                              void* d_out, int out_size, void* d_ws, size_t ws_size,
                              hipStream_t stream)
{
    const float*     points   = (const float*)d_in[0];      // [B][P][3] f32
    const float*     tfcolor  = (const float*)d_in[1];      // [B][P][3] f32
    const long long* faces    = (const long long*)d_in[2];  // [1][F][3] i64
    const float*     cam_rot  = (const float*)d_in[3];      // [B][3][3] f32
    const float*     cam_pos  = (const float*)d_in[4];      // [B][3]    f32
    const float*     cam_proj = (const float*)d_in[5];      // [1][3][1] f32
    float*           out      = (float*)d_out;

    const int B = in_sizes[4] / 3;            // 16
    const int P = in_sizes[0] / (3 * B);      // 100000
    const int F = in_sizes[2] / 3;            // 200000

    float* pts4 = (float*)d_ws;               // B*P*4 floats = 25.6 MB

    // Phase 1: WMMA point transform. 8 waves/block, 16 points per wave.
    dim3 g1((P + 127) / 128, B);
    transform_points_wmma<<<g1, 256, 0, stream>>>(points, cam_rot, cam_pos,
                                                  cam_proj, pts4, P);

    // Phase 2: face assembly, NT streaming output.
    dim3 g2((F + 255) / 256);
    assemble_faces<<<g2, 256, 0, stream>>>(pts4, tfcolor, faces, cam_proj,
                                           out, P, F, B);
}